// DyHuCoG_30039001268245
// MI455X (gfx1250) — compile-verified
//
#include <hip/hip_runtime.h>
#include <hip/hip_bf16.h>

// LightGCN propagation on MI455X (gfx1250, wave32).
// out = (emb + A*emb + A*(A*emb)) / 3 with A in COO form (6.4M edges, D=32).
// Memory-bound scatter SpMM: ~3.5 GB moved, 0.82 GFLOP -> 0.23 FLOP/B.
// Working set (~105 MB) fits in 192 MB L2, so gathers/atomics resolve in L2.
// Lane l of each wave32 owns latent dim l (D == 32 == warpSize).
// Edge metadata is staged into LDS via the gfx1250 async-DMA path
// (global_load_async_to_lds_b32 / ASYNCcnt), then broadcast to SGPRs via
// v_readlane so gather/atomic addresses are saddr-form (scalar base + lane).

#define LATENT 32
#define WAVES_PER_BLOCK 8

__device__ __forceinline__ float readlane_f32(float v, int lane) {
  return __int_as_float(__builtin_amdgcn_readlane(__float_as_int(v), lane));
}

__global__ __launch_bounds__(256) void dyhucog_zero2(
    float4* __restrict__ t1, float4* __restrict__ out, int n4) {
  int i = blockIdx.x * blockDim.x + threadIdx.x;
  if (i < n4) {
    float4 z = {0.0f, 0.0f, 0.0f, 0.0f};
    t1[i]  = z;
    out[i] = z;
  }
}

// y[row[e]][lane] += val[e] * x[col[e]][lane], one wave per 32-edge batch.
__global__ __launch_bounds__(256) void dyhucog_spmm(
    const float* __restrict__ x,
    const float* __restrict__ vals,
    const int*   __restrict__ rows,
    const int*   __restrict__ cols,
    float*       __restrict__ y,
    int nEdges) {
  // [wave][array: 0=val 1=row 2=col][edge-in-batch]
  __shared__ int meta[WAVES_PER_BLOCK][3][32];

  const int lane = threadIdx.x & 31;
  const int wib  = (threadIdx.x >> 5) & (WAVES_PER_BLOCK - 1);
  const int wave = (blockIdx.x * blockDim.x + threadIdx.x) >> 5;
  const int base = wave << 5;               // first edge of this wave's batch
  if (base >= nEdges) return;

  if (base + 32 <= nEdges) {
    // ---- Full batch (the only path for E = 6.4M) ----
    // Stage 3 x 128 B of edge metadata into LDS with the async DMA path:
    // no VGPR return, tracked on ASYNCcnt instead of LOADcnt.
    {
      const unsigned goff = (unsigned)(base + lane) * 4u;   // byte offset
      const unsigned lv = (unsigned)(uintptr_t)&meta[wib][0][lane];
      const unsigned lr = (unsigned)(uintptr_t)&meta[wib][1][lane];
      const unsigned lc = (unsigned)(uintptr_t)&meta[wib][2][lane];
      asm volatile("global_load_async_to_lds_b32 %0, %1, %2"
                   :: "v"(lv), "v"(goff), "s"(vals) : "memory");
      asm volatile("global_load_async_to_lds_b32 %0, %1, %2"
                   :: "v"(lr), "v"(goff), "s"(rows) : "memory");
      asm volatile("global_load_async_to_lds_b32 %0, %1, %2"
                   :: "v"(lc), "v"(goff), "s"(cols) : "memory");
      asm volatile("s_wait_asynccnt 0x0" ::: "memory");
    }
    const float v = __int_as_float(meta[wib][0][lane]);  // ds_load_b32, stride-4
    const int   r = meta[wib][1][lane];
    const int   c = meta[wib][2][lane];

    // Prime prefetch for edge 0's gather row (two 64B halves).
    {
      int c0 = __builtin_amdgcn_readlane(c, 0);
      if ((lane & 15) == 0)
        __builtin_prefetch(x + (size_t)c0 * LATENT + lane, 0, 3); // global_prefetch_b8
    }
#pragma unroll
    for (int k = 0; k < 32; ++k) {
      const int   ck = __builtin_amdgcn_readlane(c, k);   // SGPR (uniform)
      const int   rk = __builtin_amdgcn_readlane(r, k);   // SGPR (uniform)
      const float vk = readlane_f32(v, k);                // SGPR (uniform)

      if (k + 1 < 32) {
        int cn = __builtin_amdgcn_readlane(c, k + 1);
        if ((lane & 15) == 0)
          __builtin_prefetch(x + (size_t)cn * LATENT + lane, 0, 3);
      }

      const float xv = x[(size_t)ck * LATENT + lane];     // saddr load, L2-hot
      atomicAdd(&y[(size_t)rk * LATENT + lane], vk * xv); // global_atomic_add_f32
    }
  } else {
    // ---- Partial tail batch: direct loads + dynamic shfl broadcast ----
    const int e = base + lane;
    float v = 0.0f;
    int   r = 0, c = 0;
    if (e < nEdges) {
      v = __builtin_nontemporal_load(vals + e);
      r = __builtin_nontemporal_load(rows + e);
      c = __builtin_nontemporal_load(cols + e);
    }
    const int count = nEdges - base;
    for (int k = 0; k < count; ++k) {
      const float vk = __shfl(v, k);
      const int   rk = __shfl(r, k);
      const int   ck = __shfl(c, k);
      const float xv = x[(size_t)ck * LATENT + lane];
      atomicAdd(&y[(size_t)rk * LATENT + lane], vk * xv);
    }
  }
}

// out = (emb + t1 + out) / 3, vectorized B128.
__global__ __launch_bounds__(256) void dyhucog_combine(
    const float4* __restrict__ emb,
    const float4* __restrict__ t1,
    float4*       __restrict__ out, int n4) {
  int i = blockIdx.x * blockDim.x + threadIdx.x;
  if (i < n4) {
    float4 a = emb[i], b = t1[i], o = out[i];
    const float s = 1.0f / 3.0f;
    o.x = (a.x + b.x + o.x) * s;
    o.y = (a.y + b.y + o.y) * s;
    o.z = (a.z + b.z + o.z) * s;
    o.w = (a.w + b.w + o.w) * s;
    out[i] = o;
  }
}

extern "C" void kernel_launch(void* const* d_in, const int* in_sizes, int n_in,
                              void* d_out, int out_size, void* d_ws, size_t ws_size,
                              hipStream_t stream) {
  const float* emb  = (const float*)d_in[0]; // [N_NODES*32]
  const float* vals = (const float*)d_in[1]; // [N_EDGES]
  const int*   rows = (const int*)  d_in[2]; // [N_EDGES]
  const int*   cols = (const int*)  d_in[3]; // [N_EDGES]
  float*       out  = (float*)d_out;         // [N_NODES*32]

  const int nEdges = in_sizes[1];
  const int nElem  = in_sizes[0];            // N_NODES * 32 = 3200032 (div by 4)
  const int n4     = nElem / 4;

  float* t1 = (float*)d_ws;                  // layer-1 embedding, 12.8 MB

  // 1) zero t1 and out (deterministic per call; d_out is accumulated into).
  {
    int blocks = (n4 + 255) / 256;
    dyhucog_zero2<<<blocks, 256, 0, stream>>>((float4*)t1, (float4*)out, n4);
  }

  // 2) t1 = A * emb
  {
    int waves  = (nEdges + 31) / 32;
    int blocks = (waves + WAVES_PER_BLOCK - 1) / WAVES_PER_BLOCK;
    dyhucog_spmm<<<blocks, 256, 0, stream>>>(emb, vals, rows, cols, t1, nEdges);
  }

  // 3) out = A * t1   (accumulated via atomics into zeroed d_out)
  {
    int waves  = (nEdges + 31) / 32;
    int blocks = (waves + WAVES_PER_BLOCK - 1) / WAVES_PER_BLOCK;
    dyhucog_spmm<<<blocks, 256, 0, stream>>>(t1, vals, rows, cols, out, nEdges);
  }

  // 4) out = (emb + t1 + out) / 3
  {
    int blocks = (n4 + 255) / 256;
    dyhucog_combine<<<blocks, 256, 0, stream>>>((const float4*)emb, (const float4*)t1,
                                                (float4*)out, n4);
  }
}